// Channelenhance_65146063945877
// MI455X (gfx1250) — compile-verified
//
#include <hip/hip_runtime.h>
#include <hip/hip_bf16.h>

// MI455X / gfx1250, wave32. Problem is bandwidth-bound (~768 MiB traffic @ 23.3 TB/s);
// the tiny MLP is run on the matrix core with V_WMMA_F32_16X16X4_F32.

typedef float v2f __attribute__((ext_vector_type(2)));
typedef float v4f __attribute__((ext_vector_type(4)));
typedef float v8f __attribute__((ext_vector_type(8)));

#define N_  16
#define C_  256
#define CH  128
#define HW  16384   // 128*128

// ---------------------------------------------------------------------------
// Kernel 1: z[n][c] = mean over the 128x128 plane. One block per (n,c) plane.
// 256 threads x 16 float4 = 16384 floats, coalesced, nontemporal (no reuse).
// ---------------------------------------------------------------------------
__global__ void mean_kernel(const float* __restrict__ x, float* __restrict__ z) {
    __shared__ float red[256];
    const int b = blockIdx.x;                       // b = n*256 + c
    const v4f* src = (const v4f*)(x + (size_t)b * HW);
    float sum = 0.0f;
#pragma unroll
    for (int i = 0; i < 16; ++i) {
        v4f v = __builtin_nontemporal_load(&src[threadIdx.x + i * 256]);
        sum += v.x + v.y + v.z + v.w;
    }
    red[threadIdx.x] = sum;
    __syncthreads();
    for (int s = 128; s > 0; s >>= 1) {
        if (threadIdx.x < s) red[threadIdx.x] += red[threadIdx.x + s];
        __syncthreads();
    }
    if (threadIdx.x == 0) z[b] = red[0] * (1.0f / (float)HW);
}

// ---------------------------------------------------------------------------
// Kernel 2: single block, 8 wave32s. WMMA fp32 MLP + stable descending ranking.
//   h      = relu(z @ W1^T + b1)        (16x256)x(256x128)
//   scores = sigmoid(h @ W2^T + b2)     (16x128)x(128x256)
//   perm[n][slot] = channel occupying output slot (0..127 selected, 128..255 rest)
//
// V_WMMA_F32_16X16X4_F32 fragment layouts (ISA 7.12.2, wave32):
//   A 16x4 : lane(0-15)=row M, lane[4] selects K-half; vgpr0=K(+0), vgpr1=K(+1)
//   B 4x16 : lane(0-15)=col N, lane[4] selects K-half; vgpr0=K(+0), vgpr1=K(+1)
//   C/D    : vgpr i -> M=i (lanes 0-15) / M=i+8 (lanes 16-31), N=lane&15
// ---------------------------------------------------------------------------
__global__ void mlp_sort_kernel(const float* __restrict__ z,
                                const float* __restrict__ W1,
                                const float* __restrict__ b1,
                                const float* __restrict__ W2,
                                const float* __restrict__ b2,
                                int* __restrict__ perm) {
    __shared__ float zl[N_][C_];   // 16 KB
    __shared__ float hl[N_][CH];   //  8 KB
    __shared__ float sl[N_][C_];   // 16 KB

    const int tid   = threadIdx.x;
    const int lane  = tid & 31;
    const int wave  = tid >> 5;            // 0..7
    const int row   = lane & 15;           // M (for A) / N (for B)
    const int khalf = (lane >> 4) * 2;     // K-half select
    const int mbase = (lane >> 4) * 8;     // C/D row base

    for (int i = tid; i < N_ * C_; i += 256) zl[i >> 8][i & 255] = z[i];
    __syncthreads();

    // ---- GEMM1: h = relu(z @ W1^T + b1); wave w owns columns [16w, 16w+16) ----
    {
        const int n0 = wave * 16;
        v8f acc = {};
        for (int k = 0; k < C_; k += 4) {
            v2f a, bf;
            a.x = zl[row][k + khalf];
            a.y = zl[row][k + khalf + 1];
            const float* w1r = W1 + (size_t)(n0 + row) * C_ + (k + khalf);
            bf.x = w1r[0];
            bf.y = w1r[1];
            acc = __builtin_amdgcn_wmma_f32_16x16x4_f32(
                false, a, false, bf, (short)0, acc, false, false);
        }
        const float bias = b1[n0 + row];
#pragma unroll
        for (int i = 0; i < 8; ++i) {
            float v = acc[i] + bias;
            hl[mbase + i][n0 + row] = v > 0.0f ? v : 0.0f;
        }
    }
    __syncthreads();

    // ---- GEMM2: scores = sigmoid(h @ W2^T + b2); 16 tiles, 2 per wave ----
    for (int t = wave; t < 16; t += 8) {       // uniform: every wave runs 2 iters
        const int n0 = t * 16;
        v8f acc = {};
        for (int k = 0; k < CH; k += 4) {
            v2f a, bf;
            a.x = hl[row][k + khalf];
            a.y = hl[row][k + khalf + 1];
            const float* w2r = W2 + (size_t)(n0 + row) * CH + (k + khalf);
            bf.x = w2r[0];
            bf.y = w2r[1];
            acc = __builtin_amdgcn_wmma_f32_16x16x4_f32(
                false, a, false, bf, (short)0, acc, false, false);
        }
        const float bias = b2[n0 + row];
#pragma unroll
        for (int i = 0; i < 8; ++i) {
            float v = acc[i] + bias;
            sl[mbase + i][n0 + row] = 1.0f / (1.0f + __expf(-v));
        }
    }
    __syncthreads();

    // ---- Stable descending ranking (matches jnp.argsort(-scores) stability) ----
    // rank(c) = #{cc : s[cc] > s[c]} + #{cc < c : s[cc] == s[c]}
    for (int p = tid; p < N_ * C_; p += 256) {
        const int n = p >> 8, c = p & 255;
        const float s = sl[n][c];
        int rank = 0;
        for (int cc = 0; cc < C_; ++cc) {
            const float s2 = sl[n][cc];
            rank += (s2 > s) || ((s2 == s) && (cc < c));
        }
        perm[n * C_ + rank] = c;   // every slot written exactly once (permutation)
    }
}

// ---------------------------------------------------------------------------
// Kernel 3: permuted plane copy. One block per output slot (n, j).
// Output = [selected (16,128,128,128) | remaining (16,128,128,128)] flat.
// Streaming: nontemporal 128-bit loads/stores (no reuse -> keep L2 clean).
// ---------------------------------------------------------------------------
__global__ void gather_kernel(const float* __restrict__ x,
                              const int* __restrict__ perm,
                              float* __restrict__ out) {
    const int b = blockIdx.x;            // n*256 + j
    const int n = b >> 8;
    const int j = b & 255;
    const int c = perm[b];               // wave-uniform -> scalar load

    const v4f* src = (const v4f*)(x + ((size_t)(n * C_ + c)) * HW);
    size_t plane = (j < CH) ? (size_t)(n * CH + j)
                            : (size_t)(N_ * CH) + (size_t)(n * CH + (j - CH));
    v4f* dst = (v4f*)(out + plane * HW);

#pragma unroll
    for (int i = 0; i < 16; ++i) {
        v4f v = __builtin_nontemporal_load(&src[threadIdx.x + i * 256]);
        __builtin_nontemporal_store(v, &dst[threadIdx.x + i * 256]);
    }
}

// ---------------------------------------------------------------------------
extern "C" void kernel_launch(void* const* d_in, const int* in_sizes, int n_in,
                              void* d_out, int out_size, void* d_ws, size_t ws_size,
                              hipStream_t stream) {
    const float* x  = (const float*)d_in[0];
    const float* W1 = (const float*)d_in[1];
    const float* b1 = (const float*)d_in[2];
    const float* W2 = (const float*)d_in[3];
    const float* b2 = (const float*)d_in[4];

    float* z  = (float*)d_ws;                                   // 16 KiB
    int* perm = (int*)((char*)d_ws + (size_t)N_ * C_ * sizeof(float)); // 16 KiB

    mean_kernel   <<<N_ * C_, 256, 0, stream>>>(x, z);
    mlp_sort_kernel<<<1,      256, 0, stream>>>(z, W1, b1, W2, b2, perm);
    gather_kernel <<<N_ * C_, 256, 0, stream>>>(x, perm, (float*)d_out);
}